// GCN_71322226917733
// MI455X (gfx1250) — compile-verified
//
#include <hip/hip_runtime.h>
#include <math.h>

// ---------------------------------------------------------------------------
// GCN (3x GCNConv improved=True + dense head) for MI455X / gfx1250.
// Dense GEMMs use V_WMMA_F32_16X16X4_F32 (full fp32 — GEMMs are ~8 GFLOP and
// not the bottleneck; the L2-resident edge scatter with f32 atomics is).
// ---------------------------------------------------------------------------

typedef float v2f __attribute__((ext_vector_type(2)));
typedef float v8f __attribute__((ext_vector_type(8)));

#define CH 128  // hidden width == label count

// ---------------- degree / normalization --------------------------------

__global__ void k_init_deg(float* deg, int n) {
  int i = blockIdx.x * blockDim.x + threadIdx.x;
  if (i < n) deg[i] = 2.0f;  // improved=True self-loop weight
}

__global__ void k_accum_deg(const long long* __restrict__ dst, float* deg, int e) {
  int i = blockIdx.x * blockDim.x + threadIdx.x;
  if (i < e) atomicAdd(&deg[(int)dst[i]], 1.0f);
}

__global__ void k_dinv(float* deg_inout, float* selfnorm, int n) {
  int i = blockIdx.x * blockDim.x + threadIdx.x;
  if (i >= n) return;
  float d = deg_inout[i];          // always >= 2
  float di = rsqrtf(d);
  deg_inout[i] = di;               // becomes dinv[]
  selfnorm[i] = 2.0f * di * di;    // dinv * 2.0 * dinv  (self-loop norm)
}

__global__ void k_edge_norm(const long long* __restrict__ src,
                            const long long* __restrict__ dst,
                            const float* __restrict__ dinv,
                            float* __restrict__ norm, int e) {
  int i = blockIdx.x * blockDim.x + threadIdx.x;
  if (i >= e) return;
  norm[i] = dinv[(int)src[i]] * dinv[(int)dst[i]];
}

// ---------------- fp32 WMMA GEMM: C[M,N] = A[M,K] * B[K,N] ---------------
// One wave per 16x16 output tile. A-frag per ISA layout: lane<16 holds
// (k0, k0+1) of row M=lane; lane>=16 holds (k0+2, k0+3) of row M=lane-16.
// B-frag mirrors it over columns. Optional fused bias+sigmoid epilogue.

__global__ void k_gemm_wmma_f32(const float* __restrict__ A,
                                const float* __restrict__ B,
                                float* __restrict__ C,
                                int M, int K, int N,
                                const float* __restrict__ bias_sig) {
  const int wave = threadIdx.x >> 5;
  const int lane = threadIdx.x & 31;
  const int wavesPerBlock = blockDim.x >> 5;
  const int tilesN = N >> 4;
  const int tilesM = M >> 4;
  int tile = blockIdx.x * wavesPerBlock + wave;
  if (tile >= tilesM * tilesN) return;  // wave-uniform exit; EXEC all-1 inside
  const int tm = tile / tilesN;
  const int tn = tile - tm * tilesN;
  const int m0 = tm << 4;
  const int n0 = tn << 4;
  const int l15 = lane & 15;
  const int half = lane >> 4;

  const float* aRow = A + (size_t)(m0 + l15) * K;  // row M = l15 (both halves)
  const float* bCol = B + (size_t)n0 + l15;        // column N = l15

  v8f acc = {};
  for (int k0 = 0; k0 < K; k0 += 4) {
    const int ka = k0 + (half << 1);
    v2f a, b;
    a.x = aRow[ka];
    a.y = aRow[ka + 1];                 // contiguous float2 per lane
    b.x = bCol[(size_t)ka * N];         // coalesced across lanes
    b.y = bCol[(size_t)(ka + 1) * N];
    acc = __builtin_amdgcn_wmma_f32_16x16x4_f32(
        /*neg_a=*/false, a, /*neg_b=*/false, b,
        /*c_mod=*/(short)0, acc, /*reuse_a=*/false, /*reuse_b=*/false);
  }

  // C/D layout: lane gives column n0+l15; VGPR j gives row m0 + j + 8*half.
  const int col = n0 + l15;
  const int rbase = m0 + (half << 3);
  float bv = bias_sig ? bias_sig[col] : 0.0f;
#pragma unroll
  for (int j = 0; j < 8; ++j) {
    float v = acc[j];
    if (bias_sig) {
      v += bv;
      v = 1.0f / (1.0f + __expf(-v));  // fused sigmoid head
    }
    C[(size_t)(rbase + j) * N + col] = v;
  }
}

// ---------------- aggregation ------------------------------------------

__global__ void k_zero4(float4* p, long long n4) {
  long long i = (long long)blockIdx.x * blockDim.x + threadIdx.x;
  if (i < n4) p[i] = make_float4(0.f, 0.f, 0.f, 0.f);
}

// One wave per edge; each lane owns 4 of the 128 channels.
__global__ void k_scatter(const long long* __restrict__ src,
                          const long long* __restrict__ dst,
                          const float* __restrict__ norm,
                          const float* __restrict__ h,
                          float* __restrict__ out, int e) {
  int w = (int)(((long long)blockIdx.x * blockDim.x + threadIdx.x) >> 5);
  int lane = threadIdx.x & 31;
  if (w >= e) return;
  const int s = (int)src[w];
  const int d = (int)dst[w];
  const float nv = norm[w];
  const float4 m = ((const float4*)(h + (size_t)s * CH))[lane];  // L2 gather
  float* o = out + (size_t)d * CH + lane * 4;
  atomicAdd(o + 0, m.x * nv);
  atomicAdd(o + 1, m.y * nv);
  atomicAdd(o + 2, m.z * nv);
  atomicAdd(o + 3, m.w * nv);
}

// out = (relu?)(out + h * selfnorm[node] + bias), 4 channels per thread.
__global__ void k_finalize(float* __restrict__ out, const float* __restrict__ h,
                           const float* __restrict__ selfnorm,
                           const float* __restrict__ bias, int n, int do_relu) {
  int idx = blockIdx.x * blockDim.x + threadIdx.x;
  const int total = n * (CH / 4);
  if (idx >= total) return;
  const int node = idx >> 5;  // CH/4 == 32 groups per node
  const int g = idx & 31;
  const float sn = selfnorm[node];
  const float4 hv = ((const float4*)(h + (size_t)node * CH))[g];
  float4 ov = ((float4*)(out + (size_t)node * CH))[g];
  const float4 bv = ((const float4*)bias)[g];
  ov.x += hv.x * sn + bv.x;
  ov.y += hv.y * sn + bv.y;
  ov.z += hv.z * sn + bv.z;
  ov.w += hv.w * sn + bv.w;
  if (do_relu) {
    ov.x = fmaxf(ov.x, 0.f);
    ov.y = fmaxf(ov.y, 0.f);
    ov.z = fmaxf(ov.z, 0.f);
    ov.w = fmaxf(ov.w, 0.f);
  }
  ((float4*)(out + (size_t)node * CH))[g] = ov;
}

// ---------------- driver -------------------------------------------------

static inline size_t align256(size_t x) { return (x + 255) & ~(size_t)255; }

extern "C" void kernel_launch(void* const* d_in, const int* in_sizes, int n_in,
                              void* d_out, int out_size, void* d_ws, size_t ws_size,
                              hipStream_t stream) {
  const float* x = (const float*)d_in[0];
  const long long* ei = (const long long*)d_in[1];  // jnp.int64 in reference
  const float* W1 = (const float*)d_in[2];
  const float* b1 = (const float*)d_in[3];
  const float* W2 = (const float*)d_in[4];
  const float* b2 = (const float*)d_in[5];
  const float* We = (const float*)d_in[6];
  const float* be = (const float*)d_in[7];
  const float* Wf = (const float*)d_in[8];
  const float* bf = (const float*)d_in[9];
  float* out = (float*)d_out;

  const int FIN = 256;
  const int n = in_sizes[0] / FIN;  // 50000 (multiple of 16)
  const int e = in_sizes[1] / 2;    // 800000
  const long long* src = ei;
  const long long* dst = ei + e;

  // Workspace carve-up (~54.8 MB total)
  char* ws = (char*)d_ws;
  size_t off = 0;
  float* deg = (float*)(ws + off);       off += align256((size_t)n * 4);   // -> dinv
  float* selfnorm = (float*)(ws + off);  off += align256((size_t)n * 4);
  float* norm = (float*)(ws + off);      off += align256((size_t)e * 4);
  float* hA = (float*)(ws + off);        off += align256((size_t)n * CH * 4);
  float* hB = (float*)(ws + off);        off += align256((size_t)n * CH * 4);
  (void)ws_size;

  const int T = 256;
  // --- normalization ---
  k_init_deg<<<(n + T - 1) / T, T, 0, stream>>>(deg, n);
  k_accum_deg<<<(e + T - 1) / T, T, 0, stream>>>(dst, deg, e);
  k_dinv<<<(n + T - 1) / T, T, 0, stream>>>(deg, selfnorm, n);
  k_edge_norm<<<(e + T - 1) / T, T, 0, stream>>>(src, dst, deg, norm, e);

  const long long n4 = (long long)n * CH / 4;
  const int zb = (int)((n4 + T - 1) / T);
  const int fb = (n * (CH / 4) + T - 1) / T;
  const int sb = (e * 32 + T - 1) / T;  // wave per edge, 8 waves/block

  auto gemm = [&](const float* A, const float* B, float* C, int M, int K, int N,
                  const float* bias_sig) {
    int tiles = (M >> 4) * (N >> 4);
    int blocks = (tiles + 3) / 4;  // 4 waves (16x16 tiles) per 128-thread block
    k_gemm_wmma_f32<<<blocks, 128, 0, stream>>>(A, B, C, M, K, N, bias_sig);
  };

  // --- layer 1: relu(agg(x @ W1) + b1) ---
  gemm(x, W1, hA, n, FIN, CH, nullptr);
  k_zero4<<<zb, T, 0, stream>>>((float4*)hB, n4);
  k_scatter<<<sb, T, 0, stream>>>(src, dst, norm, hA, hB, e);
  k_finalize<<<fb, T, 0, stream>>>(hB, hA, selfnorm, b1, n, 1);

  // --- layer 2: relu(agg(h @ W2) + b2) ---
  gemm(hB, W2, hA, n, CH, CH, nullptr);
  k_zero4<<<zb, T, 0, stream>>>((float4*)hB, n4);
  k_scatter<<<sb, T, 0, stream>>>(src, dst, norm, hA, hB, e);
  k_finalize<<<fb, T, 0, stream>>>(hB, hA, selfnorm, b2, n, 1);

  // --- layer 3: agg(h @ We) + be (no relu) ---
  gemm(hB, We, hA, n, CH, CH, nullptr);
  k_zero4<<<zb, T, 0, stream>>>((float4*)hB, n4);
  k_scatter<<<sb, T, 0, stream>>>(src, dst, norm, hA, hB, e);
  k_finalize<<<fb, T, 0, stream>>>(hB, hA, selfnorm, be, n, 0);

  // --- head: sigmoid(h @ Wf + bf) -> d_out ---
  gemm(hB, Wf, out, n, CH, CH, bf);
  (void)out_size; (void)n_in;
}